// MultiHeadAttentionBlock_23665269801423
// MI455X (gfx1250) — compile-verified
//
#include <hip/hip_runtime.h>

// ---- problem constants ----
constexpr int Bc = 2, Sc = 2048, Dc = 512, Hc = 8, DHc = 64, Pc = 10;
constexpr int Tc = 2 * Pc + 1;               // 21 relative bins
#define SCALE_F 22.627416997969522f          // sqrt(512)
#define INV_SCALE (1.0f / SCALE_F)

typedef __attribute__((ext_vector_type(2))) float v2f;
typedef __attribute__((ext_vector_type(8))) float v8f;

// CDNA5 fp32 WMMA: D(16x16,f32) = A(16x4,f32) x B(4x16,f32) + C
// A: lane m=lane%16 holds row m; VGPR0={K=0|K=2 by half}, VGPR1={K=1|K=3}
// B: lane n=lane%16 holds col n; same K split by lane half
__device__ inline v8f wmma4(v2f a, v2f b, v8f c) {
  return __builtin_amdgcn_wmma_f32_16x16x4_f32(
      /*neg_a=*/false, a, /*neg_b=*/false, b,
      /*c_mod=*/(short)0, c, /*reuse_a=*/false, /*reuse_b=*/false);
}

__device__ inline float wsum32(float v) {
#pragma unroll
  for (int m = 16; m > 0; m >>= 1) v += __shfl_xor(v, m, 32);
  return v;
}

// gfx1250 async global->LDS copy (ASYNCcnt path, bypasses VGPRs).
// 16 bytes per lane; LDS dest offset must be 16B aligned.
__device__ inline void async_copy16(const float* gptr, float* lptr) {
  unsigned lds = (unsigned)(size_t)lptr;   // low 32 bits = LDS offset
  asm volatile("global_load_async_to_lds_b128 %0, %1, off"
               :
               : "v"(lds), "v"(gptr)
               : "memory");
}
__device__ inline void wait_async0() {
  asm volatile("s_wait_asynccnt 0" ::: "memory");
}

// ---------------------------------------------------------------------------
// Generic fp32 WMMA GEMM: out = A[4096x512] * W[512x512] + bias[512]
// Block = 256 thr (8 waves). Block tile: 128 rows x 64 cols. Wave: 16x64 strip.
// A panels staged via GLOBAL_LOAD_ASYNC_TO_LDS_B128 (no VGPR round-trip).
// headMajor: scatter out[b,s, col] -> [B,H,S,DH] head-major layout.
// ---------------------------------------------------------------------------
#define APAD 36  // row stride 144B: 16B-aligned; stride-36 mod 64 banks -> 16
                 // distinct banks for the 16-lane column access (conflict-free)
__global__ void gemm_wmma_f32(const float* __restrict__ A,
                              const float* __restrict__ W,
                              const float* __restrict__ bias,
                              float* __restrict__ out, int headMajor) {
  __shared__ float As[128 * APAD];
  const int tid = threadIdx.x;
  const int w = tid >> 5, lane = tid & 31;
  const int m = lane & 15, half = lane >> 4;
  const int row_base = (blockIdx.x & 31) * 128;
  const int col0 = (blockIdx.x >> 5) * 64;

  v8f c[4] = {};

  for (int k0 = 0; k0 < Dc; k0 += 32) {
    __syncthreads();  // panel consumed by all waves before overwrite
    // async-stage 128x32 fp32 A panel: 4x B128 per thread, coalesced
#pragma unroll
    for (int i = 0; i < 4; i++) {
      int idx = tid + i * 256;            // 1024 x 16B = 128*32 floats
      int rr = idx >> 3;
      int cc = (idx & 7) * 4;
      async_copy16(A + (size_t)(row_base + rr) * Dc + k0 + cc,
                   &As[rr * APAD + cc]);
    }
    wait_async0();   // own async writes to LDS complete
    __syncthreads(); // all waves' panel slices visible

#pragma unroll
    for (int kk = 0; kk < 32; kk += 4) {
      const int kl = kk + 2 * half;
      v2f a;
      a.x = As[(w * 16 + m) * APAD + kl];
      a.y = As[(w * 16 + m) * APAD + kl + 1];
#pragma unroll
      for (int j = 0; j < 4; j++) {
        const int n = col0 + j * 16 + m;
        v2f bv;
        bv.x = W[(size_t)(k0 + kl) * Dc + n];
        bv.y = W[(size_t)(k0 + kl + 1) * Dc + n];
        c[j] = wmma4(a, bv, c[j]);
      }
    }
  }

#pragma unroll
  for (int j = 0; j < 4; j++) {
#pragma unroll
    for (int rr = 0; rr < 8; rr++) {
      const int row = row_base + w * 16 + rr + 8 * half;  // C: M = rr + 8*half
      const int col = col0 + j * 16 + m;                  //    N = lane%16
      const float val = c[j][rr] + bias[col];
      if (headMajor) {
        const int b = row >> 11;        // row = b*S + s
        const int s = row & (Sc - 1);
        const int hh = col >> 6, d = col & 63;
        out[(((size_t)(b * Hc + hh)) * Sc + s) * DHc + d] = val;
      } else {
        out[(size_t)row * Dc + col] = val;
      }
    }
  }
}

// ---------------------------------------------------------------------------
// M[bh] = Kh[bh]^T * Vh[bh]  (64x64, reduce over S=2048). One block per (b,h).
// 8 waves: wave w -> e-tile (w%4)*16, d-pair (w/4)*32 (two 16x16 C tiles).
// ---------------------------------------------------------------------------
__global__ void ktv_wmma(const float* __restrict__ Kh,
                         const float* __restrict__ Vh, float* __restrict__ M) {
  const int bh = blockIdx.x;
  const int tid = threadIdx.x, w = tid >> 5, lane = tid & 31;
  const int m = lane & 15, half = lane >> 4;
  const int e0 = (w & 3) * 16, d0 = (w >> 2) * 32;
  const float* Kp = Kh + (size_t)bh * Sc * DHc;
  const float* Vp = Vh + (size_t)bh * Sc * DHc;
  v8f c0 = {}, c1 = {};
  for (int s0 = 0; s0 < Sc; s0 += 4) {
    const int k = s0 + 2 * half;
    v2f a;                                 // A[m,kl] = Kh[s0+kl, e0+m]
    a.x = Kp[(size_t)k * DHc + e0 + m];
    a.y = Kp[(size_t)(k + 1) * DHc + e0 + m];
    v2f b0v, b1v;                          // B[kl,n] = Vh[s0+kl, d0+n]
    b0v.x = Vp[(size_t)k * DHc + d0 + m];
    b0v.y = Vp[(size_t)(k + 1) * DHc + d0 + m];
    b1v.x = Vp[(size_t)k * DHc + d0 + 16 + m];
    b1v.y = Vp[(size_t)(k + 1) * DHc + d0 + 16 + m];
    c0 = wmma4(a, b0v, c0);
    c1 = wmma4(a, b1v, c1);
  }
  float* Mp = M + (size_t)bh * DHc * DHc;
#pragma unroll
  for (int r = 0; r < 8; r++) {
    const int e = e0 + r + 8 * half;
    Mp[e * DHc + d0 + m] = c0[r];
    Mp[e * DHc + d0 + 16 + m] = c1[r];
  }
}

// ---------------------------------------------------------------------------
// Exclusive prefix sums of Kh, Vh along S per (b,h,d): cum[s] = sum_{k<s}.
// cum has S+1 rows (cum[S] = total). 128 threads: d<64 -> K, else V.
// ---------------------------------------------------------------------------
__global__ void scan_kernel(const float* __restrict__ Kh,
                            const float* __restrict__ Vh,
                            float* __restrict__ cumK, float* __restrict__ cumV) {
  const int bh = blockIdx.x;
  const int t = threadIdx.x;
  const float* src = (t < DHc) ? Kh : Vh;
  float* dst = (t < DHc) ? cumK : cumV;
  const int d = t & (DHc - 1);
  const float* s = src + (size_t)bh * Sc * DHc + d;
  float* o = dst + (size_t)bh * (Sc + 1) * DHc + d;
  float acc = 0.f;
  for (int i = 0; i < Sc; i++) {
    o[(size_t)i * DHc] = acc;
    acc += s[(size_t)i * DHc];
  }
  o[(size_t)Sc * DHc] = acc;
}

// ---------------------------------------------------------------------------
// TermA: X[b,s,h*64+d] = (1/sqrt(D)) * (Qh[bh] @ M[bh])  (batched WMMA GEMM)
// ---------------------------------------------------------------------------
__global__ void terma_wmma(const float* __restrict__ Qh,
                           const float* __restrict__ M, float* __restrict__ X) {
  const int bh = blockIdx.y, b = bh >> 3, h = bh & 7;
  const int tid = threadIdx.x, w = tid >> 5, lane = tid & 31;
  const int m = lane & 15, half = lane >> 4;
  const int q0 = (blockIdx.x * 8 + w) * 16;
  const float* Qp = Qh + (size_t)bh * Sc * DHc;
  const float* Mp = M + (size_t)bh * DHc * DHc;
  v8f c[4] = {};
  for (int k0 = 0; k0 < DHc; k0 += 4) {
    const int k = k0 + 2 * half;
    v2f a;
    a.x = Qp[(size_t)(q0 + m) * DHc + k];
    a.y = Qp[(size_t)(q0 + m) * DHc + k + 1];
#pragma unroll
    for (int j = 0; j < 4; j++) {
      v2f bv;
      bv.x = Mp[k * DHc + j * 16 + m];
      bv.y = Mp[(k + 1) * DHc + j * 16 + m];
      c[j] = wmma4(a, bv, c[j]);
    }
  }
#pragma unroll
  for (int j = 0; j < 4; j++)
#pragma unroll
    for (int r = 0; r < 8; r++) {
      const int qq = q0 + r + 8 * half;
      const int d = j * 16 + m;
      X[((size_t)b * Sc + qq) * Dc + h * DHc + d] = INV_SCALE * c[j][r];
    }
}

// ---------------------------------------------------------------------------
// Combine: adds relative-position terms (TermB: Rel@V, TermC: binned-scores @
// rel_v_table) to X. One wave per query row (8 q per wave). wave32 reductions.
// ---------------------------------------------------------------------------
__global__ void combine_kernel(const float* __restrict__ Qh,
                               const float* __restrict__ Kh,
                               const float* __restrict__ Vh,
                               const float* __restrict__ cumK,
                               const float* __restrict__ cumV,
                               const float* __restrict__ relK,
                               const float* __restrict__ relV,
                               float* __restrict__ X) {
  const int bh = blockIdx.y, b = bh >> 3, h = bh & 7;
  const int tid = threadIdx.x, w = tid >> 5, lane = tid & 31;
  const int d0 = lane, d1 = lane + 32;
  const float* Qp = Qh + (size_t)bh * Sc * DHc;
  const float* Kp = Kh + (size_t)bh * Sc * DHc;
  const float* Vp = Vh + (size_t)bh * Sc * DHc;
  const float* cK = cumK + (size_t)bh * (Sc + 1) * DHc;
  const float* cV = cumV + (size_t)bh * (Sc + 1) * DHc;

  for (int qi = 0; qi < 8; qi++) {
    const int q = blockIdx.x * 64 + w * 8 + qi;
    const float qv0 = Qp[(size_t)q * DHc + d0];
    const float qv1 = Qp[(size_t)q * DHc + d1];

    // R[t] = Qh[q] . relK[t]   (broadcast to all lanes via xor-reduction)
    float r[Tc];
#pragma unroll
    for (int t = 0; t < Tc; t++)
      r[t] = wsum32(qv0 * relK[t * DHc + d0] + qv1 * relK[t * DHc + d1]);

    // edge bins: t=0 -> k <= q-P ; t=2P -> k >= q+P
    int e0 = q - Pc + 1; e0 = e0 < 0 ? 0 : (e0 > Sc ? Sc : e0);
    int e1 = q + Pc;     e1 = e1 > Sc ? Sc : e1;
    const float cnt0 = (float)e0, cntT = (float)(Sc - e1);

    const float g0 = wsum32(qv0 * cK[(size_t)e0 * DHc + d0] +
                            qv1 * cK[(size_t)e0 * DHc + d1]);
    const float gT = wsum32(
        qv0 * (cK[(size_t)Sc * DHc + d0] - cK[(size_t)e1 * DHc + d0]) +
        qv1 * (cK[(size_t)Sc * DHc + d1] - cK[(size_t)e1 * DHc + d1]));

    float s21[Tc];
    s21[0] = (g0 + r[0] * cnt0) * INV_SCALE;
    s21[Tc - 1] = (gT + r[Tc - 1] * cntT) * INV_SCALE;

    float out0 = 0.f, out1 = 0.f;
#pragma unroll
    for (int j = -(Pc - 1); j <= (Pc - 1); j++) {   // middle bins: single k
      const int k = q + j, t = j + Pc;
      if (k >= 0 && k < Sc) {
        const float g = wsum32(qv0 * Kp[(size_t)k * DHc + d0] +
                               qv1 * Kp[(size_t)k * DHc + d1]);
        s21[t] = (g + r[t]) * INV_SCALE;
        out0 += INV_SCALE * r[t] * Vp[(size_t)k * DHc + d0];  // TermB taps
        out1 += INV_SCALE * r[t] * Vp[(size_t)k * DHc + d1];
      } else {
        s21[t] = 0.f;
      }
    }
    // TermB edge bins via prefix sums of Vh
    out0 += INV_SCALE * (r[0] * cV[(size_t)e0 * DHc + d0] +
                         r[Tc - 1] * (cV[(size_t)Sc * DHc + d0] -
                                      cV[(size_t)e1 * DHc + d0]));
    out1 += INV_SCALE * (r[0] * cV[(size_t)e0 * DHc + d1] +
                         r[Tc - 1] * (cV[(size_t)Sc * DHc + d1] -
                                      cV[(size_t)e1 * DHc + d1]));
    // TermC: binned scores @ rel_v_table
#pragma unroll
    for (int t = 0; t < Tc; t++) {
      out0 += s21[t] * relV[t * DHc + d0];
      out1 += s21[t] * relV[t * DHc + d1];
    }
    const size_t xi = ((size_t)b * Sc + q) * Dc + h * DHc;
    X[xi + d0] += out0;
    X[xi + d1] += out1;
  }
}

// ---------------------------------------------------------------------------
extern "C" void kernel_launch(void* const* d_in, const int* in_sizes, int n_in,
                              void* d_out, int out_size, void* d_ws,
                              size_t ws_size, hipStream_t stream) {
  const float* q = (const float*)d_in[0];
  const float* k = (const float*)d_in[1];
  const float* v = (const float*)d_in[2];
  // d_in[3] = mask (no-op in reference)
  const float* Wq = (const float*)d_in[4];
  const float* bq = (const float*)d_in[5];
  const float* Wk = (const float*)d_in[6];
  const float* bk = (const float*)d_in[7];
  const float* Wv = (const float*)d_in[8];
  const float* bv = (const float*)d_in[9];
  const float* W0 = (const float*)d_in[10];
  const float* b0 = (const float*)d_in[11];
  const float* relK = (const float*)d_in[12];
  const float* relV = (const float*)d_in[13];
  float* out = (float*)d_out;

  const size_t NBH = (size_t)Bc * Hc * Sc * DHc;       // 2,097,152
  const size_t NC = (size_t)Bc * Hc * (Sc + 1) * DHc;  // 2,098,176
  float* ws = (float*)d_ws;
  float* Qh = ws;
  float* Kh = Qh + NBH;
  float* Vh = Kh + NBH;
  float* cumK = Vh + NBH;
  float* cumV = cumK + NC;
  float* M = cumV + NC;                                // 16*64*64
  float* X = M + (size_t)Bc * Hc * DHc * DHc;          // [B,S,D]

  // 1) projections -> head-major [B,H,S,DH]
  gemm_wmma_f32<<<256, 256, 0, stream>>>(q, Wq, bq, Qh, 1);
  gemm_wmma_f32<<<256, 256, 0, stream>>>(k, Wk, bk, Kh, 1);
  gemm_wmma_f32<<<256, 256, 0, stream>>>(v, Wv, bv, Vh, 1);
  // 2) M = Kh^T Vh per (b,h)
  ktv_wmma<<<Bc * Hc, 256, 0, stream>>>(Kh, Vh, M);
  // 3) prefix sums along S
  scan_kernel<<<Bc * Hc, 128, 0, stream>>>(Kh, Vh, cumK, cumV);
  // 4) TermA = Qh @ M / sqrt(D) -> X
  terma_wmma<<<dim3(16, Bc * Hc), 256, 0, stream>>>(Qh, M, X);
  // 5) relative-position terms added to X
  combine_kernel<<<dim3(Sc / 64, Bc * Hc), 256, 0, stream>>>(
      Qh, Kh, Vh, cumK, cumV, relK, relV, X);
  // 6) output projection
  gemm_wmma_f32<<<256, 256, 0, stream>>>(X, W0, b0, out, 0);
}